// GCN_4148938408753
// MI455X (gfx1250) — compile-verified
//
#include <hip/hip_runtime.h>

typedef __attribute__((ext_vector_type(2))) float v2f;
typedef __attribute__((ext_vector_type(8))) float v8f;

constexpr int KDIM = 128;   // inner dim of both GEMMs (IN == HID == 128)

// ---------------------------------------------------------------- degrees ---
__global__ void __launch_bounds__(256) k_deg_init(float* deg, int n) {
    int i = blockIdx.x * 256 + threadIdx.x;
    if (i < n) deg[i] = 1.0f;                      // self loop contributes 1
}

__global__ void __launch_bounds__(256) k_deg_edges(const int* __restrict__ dst,
                                                   float* deg, int e) {
    int i = blockIdx.x * 256 + threadIdx.x;
    if (i < e) atomicAdd(&deg[dst[i]], 1.0f);
}

__global__ void __launch_bounds__(256) k_dinv(float* deg, int n) {
    int i = blockIdx.x * 256 + threadIdx.x;
    if (i < n) deg[i] = rsqrtf(deg[i]);            // deg >= 1 always (self loops)
}

// ------------------------------------------------------------- WMMA GEMM ----
// H[nrows x OUTC] = A[nrows x 128] @ W[128 x OUTC]     (fp32 WMMA 16x16x4)
// BIASRELU: A is transformed relu(A + bias[col]) while staging into LDS
// (fuses layer-1 bias+ReLU into the layer-2 GEMM operand load).
//
// Block = 256 threads = 8 waves, owns a 64-row stripe of A in LDS.
// Tiles: 4 row-subtiles x (OUTC/16) col-tiles; each wave keeps one B fragment
// in registers and sweeps TPW row-subtiles with it (B register reuse).
template<int OUTC, bool BIASRELU>
__global__ void __launch_bounds__(256)
k_gemm_wmma(const float* __restrict__ A, const float* __restrict__ W,
            const float* __restrict__ bias, float* __restrict__ H, int nrows)
{
    constexpr int ROWS = 64;
    __shared__ float lds[ROWS][KDIM + 4];          // +4 pad: conflict-free col reads,
                                                   // row stride 528 B keeps f4 align
    const int row0 = blockIdx.x * ROWS;
    const int tid  = threadIdx.x;

    // Stage 64x128 A stripe with float4 loads/stores (2048 f4, 8 per thread)
    for (int i = tid; i < ROWS * (KDIM / 4); i += 256) {
        const int r  = i >> 5;                     // / (KDIM/4)
        const int c4 = i & 31;
        const int row = row0 + r;
        float4 v = make_float4(0.f, 0.f, 0.f, 0.f);
        if (row < nrows) v = ((const float4*)(A + (size_t)row * KDIM))[c4];
        if (BIASRELU) {
            const float4 b4 = ((const float4*)bias)[c4];
            v.x = fmaxf(v.x + b4.x, 0.f);
            v.y = fmaxf(v.y + b4.y, 0.f);
            v.z = fmaxf(v.z + b4.z, 0.f);
            v.w = fmaxf(v.w + b4.w, 0.f);
        }
        *(float4*)&lds[r][c4 * 4] = v;
    }
    __syncthreads();

    const int wave = tid >> 5;
    const int lane = tid & 31;
    const int half = lane >> 4;                    // 0: lanes 0-15, 1: lanes 16-31
    const int l16  = lane & 15;

    constexpr int CT  = OUTC / 16;                 // col tiles (8 or 4)
    constexpr int TPW = (4 * CT) / 8;              // row-subtiles per wave (4 or 2)
    const int col0  = (wave % CT) * 16;
    const int rbase = (wave / CT) * TPW;

    v8f acc[TPW] = {};
    for (int kk = 0; kk < KDIM; kk += 4) {
        // B fragment (4x16): lane n=l16 holds W[ka][n], W[ka+1][n];
        // lanes 0-15 get K=kk,kk+1; lanes 16-31 get K=kk+2,kk+3 (ISA layout)
        const int ka = kk + half * 2;
        v2f b;
        const float* wp = W + (size_t)ka * OUTC + col0 + l16;
        b.x = wp[0];
        b.y = wp[OUTC];
#pragma unroll
        for (int t = 0; t < TPW; ++t) {
            const int rr = (rbase + t) * 16 + l16; // A row within stripe
            v2f a;
            a.x = lds[rr][ka];
            a.y = lds[rr][ka + 1];
            acc[t] = __builtin_amdgcn_wmma_f32_16x16x4_f32(
                false, a, false, b, (short)0, acc[t], false, false);
        }
    }

    // C/D layout: VGPR r -> row r (lanes 0-15) / r+8 (lanes 16-31), col = l16
#pragma unroll
    for (int t = 0; t < TPW; ++t) {
        const int rt0 = row0 + (rbase + t) * 16;
        float* outp = H + (size_t)rt0 * OUTC + col0;
        if (rt0 + 16 <= nrows) {                   // wave-uniform fast path
#pragma unroll
            for (int r = 0; r < 8; ++r)
                outp[(size_t)(r + half * 8) * OUTC + l16] = acc[t][r];
        } else {
#pragma unroll
            for (int r = 0; r < 8; ++r)
                if (rt0 + r + half * 8 < nrows)
                    outp[(size_t)(r + half * 8) * OUTC + l16] = acc[t][r];
        }
    }
}

// ------------------------------------------------- self-loop init (+bias) ---
// agg[i] = h[i] * dinv[node]^2 (+ bias[col]) ; COLS is a power of two
template<int COLS>
__global__ void __launch_bounds__(256)
k_self_init(const float* __restrict__ h, const float* __restrict__ dinv,
            const float* __restrict__ bias, float* __restrict__ agg, int total)
{
    int i = blockIdx.x * 256 + threadIdx.x;
    if (i >= total) return;
    const int node = i / COLS;                     // constexpr -> shift
    const int col  = i % COLS;
    const float d = dinv[node];
    float v = h[i] * d * d;
    if (bias) v += bias[col];
    agg[i] = v;
}

// ---------------------------------------------------------- edge scatter ----
// One wave per edge; each lane gathers COLS/32 consecutive floats (vector
// load) and issues f32 atomic adds — fully coalesced across the wave.
template<int COLS>
__global__ void __launch_bounds__(256)
k_scatter(const int* __restrict__ src, const int* __restrict__ dst,
          const float* __restrict__ dinv, const float* __restrict__ h,
          float* __restrict__ agg, int E)
{
    const int t = blockIdx.x * 256 + threadIdx.x;
    const int e = t >> 5;
    if (e >= E) return;
    const int lane = t & 31;
    const int s = src[e], d = dst[e];
    const float nrm = dinv[s] * dinv[d];

    constexpr int V = COLS / 32;                   // 4 (HID) or 2 (OUT)
    const float* hp = h   + (size_t)s * COLS + lane * V;
    float*       ap = agg + (size_t)d * COLS + lane * V;
    if constexpr (V == 4) {
        const float4 v = *(const float4*)hp;
        atomicAdd(ap + 0, v.x * nrm);
        atomicAdd(ap + 1, v.y * nrm);
        atomicAdd(ap + 2, v.z * nrm);
        atomicAdd(ap + 3, v.w * nrm);
    } else {
        const float2 v = *(const float2*)hp;
        atomicAdd(ap + 0, v.x * nrm);
        atomicAdd(ap + 1, v.y * nrm);
    }
}

// ------------------------------------------------------------------ entry ---
extern "C" void kernel_launch(void* const* d_in, const int* in_sizes, int n_in,
                              void* d_out, int out_size, void* d_ws, size_t ws_size,
                              hipStream_t stream)
{
    const float* x  = (const float*)d_in[0];
    const int*   ei = (const int*)  d_in[1];
    const float* W1 = (const float*)d_in[2];
    const float* b1 = (const float*)d_in[3];
    const float* W2 = (const float*)d_in[4];
    const float* b2 = (const float*)d_in[5];
    float* out = (float*)d_out;

    constexpr int HID = 128, OUT = 64;
    const int N = in_sizes[0] / KDIM;              // 100000
    const int E = in_sizes[1] / 2;                 // 1600000
    const int* src = ei;
    const int* dst = ei + E;

    // workspace carve-out (256B aligned)
    size_t off = 0;
    auto carve = [&](size_t bytes) {
        size_t o = off;
        off = (off + bytes + 255) & ~(size_t)255;
        return o;
    };
    char* ws = (char*)d_ws;
    float* dinv = (float*)(ws + carve((size_t)N * 4));
    float* h    = (float*)(ws + carve((size_t)N * HID * 4)); // pre-agg; reused for layer 2
    float* agg1 = (float*)(ws + carve((size_t)N * HID * 4));

    const int gemmBlocks = (N + 63) / 64;
    const int scatBlocks = (int)(((long)E * 32 + 255) / 256);

    // degrees -> dinv
    k_deg_init <<<(N + 255) / 256, 256, 0, stream>>>(dinv, N);
    k_deg_edges<<<(E + 255) / 256, 256, 0, stream>>>(dst, dinv, E);
    k_dinv     <<<(N + 255) / 256, 256, 0, stream>>>(dinv, N);

    // layer 1: h = x @ W1 ; agg1 = self + edge scatter (b1+ReLU fused into GEMM2 load)
    k_gemm_wmma<HID, false><<<gemmBlocks, 256, 0, stream>>>(x, W1, nullptr, h, N);
    k_self_init<HID><<<((N * HID) + 255) / 256, 256, 0, stream>>>(h, dinv, nullptr, agg1, N * HID);
    k_scatter<HID><<<scatBlocks, 256, 0, stream>>>(src, dst, dinv, h, agg1, E);

    // layer 2: h2 = relu(agg1 + b1) @ W2 ; out = self(+b2) + edge scatter
    k_gemm_wmma<OUT, true><<<gemmBlocks, 256, 0, stream>>>(agg1, W2, b1, h, N);
    k_self_init<OUT><<<((N * OUT) + 255) / 256, 256, 0, stream>>>(h, dinv, b2, out, N * OUT);
    k_scatter<OUT><<<scatBlocks, 256, 0, stream>>>(src, dst, dinv, h, out, E);
}